// GRU2dLayer_90177133346848
// MI455X (gfx1250) — compile-verified
//
#include <hip/hip_runtime.h>
#include <hip/hip_bf16.h>

typedef __attribute__((ext_vector_type(2))) float v2f;
typedef __attribute__((ext_vector_type(8))) float v8f;

constexpr int Bc = 8, T0c = 64, T1c = 64, Ec = 512, Hc = 512, Gc = 2048;
constexpr int ROWS = 16;       // rows per workgroup
constexpr int NWAVE = 16;      // waves per workgroup (512 threads)
constexpr float LN_EPS = 1e-5f;

__device__ __forceinline__ int iclampi(int v, int lo, int hi) {
    return v < lo ? lo : (v > hi ? hi : v);
}
__device__ __forceinline__ float sigm(float x) { return 1.0f / (1.0f + __expf(-x)); }
__device__ __forceinline__ float halfReduce(float v) {
    // reduce across each 16-lane half (rows split lo/hi by half)
    v += __shfl_xor(v, 1, 32);
    v += __shfl_xor(v, 2, 32);
    v += __shfl_xor(v, 4, 32);
    v += __shfl_xor(v, 8, 32);
    return v;
}

__global__ void gru2d_zero(float* out, size_t nOut, float* S, size_t nS) {
    size_t i = (size_t)blockIdx.x * blockDim.x + threadIdx.x;
    size_t stride = (size_t)gridDim.x * blockDim.x;
    for (size_t k = i; k < nOut; k += stride) out[k] = 0.0f;
    for (size_t k = i; k < nS; k += stride) S[k] = 0.0f;
}

// ===================== Precompute: IG = LN(x . Wi^T) * g_i + b_i =====================
// One-shot, fully parallel (no recurrence). grid = 32768/16 = 2048 WGs, 512 threads.
__global__ void __launch_bounds__(512)
gru2d_igates(const float* __restrict__ x, const float* __restrict__ Wi,
             const float* __restrict__ g_i, const float* __restrict__ b_i,
             float* __restrict__ IG) {
    __shared__ float sRed[NWAVE][ROWS][2];
    __shared__ float sStats[ROWS][2];

    const int lane = threadIdx.x & 31;
    const int wv   = threadIdx.x >> 5;
    const int hi   = lane >> 4;
    const int l15  = lane & 15;
    const int kSel = hi << 1;
    const int rowBase = blockIdx.x * ROWS;
    const int colBase = wv * 128;

    const float* xBase  = x + (size_t)(rowBase + l15) * Ec;
    const float* wiRow  = Wi + (size_t)(colBase + l15) * Ec;

    v8f c[8];
#pragma unroll
    for (int t = 0; t < 8; ++t) c[t] = (v8f)(0.0f);

#pragma unroll 2
    for (int k = 0; k < Ec; k += 4) {
        __builtin_prefetch(xBase + k + 64, 0, 0);
        float2 a2 = *(const float2*)(xBase + k + kSel);
        v2f av; av.x = a2.x; av.y = a2.y;
#pragma unroll
        for (int t = 0; t < 8; ++t) {
            float2 b2 = *(const float2*)(wiRow + (size_t)t * 16 * Ec + k + kSel);
            v2f bv; bv.x = b2.x; bv.y = b2.y;
            c[t] = __builtin_amdgcn_wmma_f32_16x16x4_f32(
                false, av, false, bv, (short)0, c[t], false, false);
        }
    }

    // row-wise LN stats over 2048 cols
#pragma unroll
    for (int r = 0; r < 8; ++r) {
        float a = 0, b = 0;
#pragma unroll
        for (int t = 0; t < 8; ++t) { float v = c[t][r]; a += v; b += v * v; }
        a = halfReduce(a); b = halfReduce(b);
        if (l15 == 0) { int row = hi * 8 + r; sRed[wv][row][0] = a; sRed[wv][row][1] = b; }
    }
    __syncthreads();
    if (threadIdx.x < ROWS) {
        int row = threadIdx.x;
        float a = 0, b = 0;
        for (int w = 0; w < NWAVE; ++w) { a += sRed[w][row][0]; b += sRed[w][row][1]; }
        float m = a / (float)Gc;
        float v = b / (float)Gc - m * m;
        sStats[row][0] = m; sStats[row][1] = __frsqrt_rn(v + LN_EPS);
    }
    __syncthreads();

#pragma unroll
    for (int t = 0; t < 8; ++t) {
        int col = colBase + 16 * t + l15;
        float gi = g_i[col], bi = b_i[col];
#pragma unroll
        for (int r = 0; r < 8; ++r) {
            int row = hi * 8 + r;
            float val = (c[t][r] - sStats[row][0]) * sStats[row][1] * gi + bi;
            IG[(size_t)(rowBase + row) * Gc + col] = val;
        }
    }
}

// ===================== One anti-diagonal step =====================
// PRE=true: igates read from IG (precomputed); PRE=false: fused igates GEMM.
template <bool PRE>
__global__ void __launch_bounds__(512)
gru2d_diag(int d,
           const float* __restrict__ x, const unsigned char* __restrict__ masks,
           const float* __restrict__ Wi, const float* __restrict__ Ws,
           const float* __restrict__ g_i, const float* __restrict__ b_i,
           const float* __restrict__ g_s, const float* __restrict__ b_s,
           const float* __restrict__ g_h, const float* __restrict__ b_h,
           const float* __restrict__ IG,
           float* __restrict__ S, float* __restrict__ outS, float* __restrict__ outF) {
    __shared__ float sG[3][ROWS][Hc];     // gate slices i/n/l   (96 KB)
    __shared__ float sSn[ROWS][Hc];       // sgates n-slice      (32 KB)
    __shared__ float sRed[NWAVE][ROWS][4];
    __shared__ float sStats[ROWS][4];     // {m1, rs1, m2, rs2}
    __shared__ float sHred[4][ROWS][2];
    __shared__ float sHstats[ROWS][2];

    const int lane = threadIdx.x & 31;
    const int wv   = threadIdx.x >> 5;     // 0..15
    const int hi   = lane >> 4;            // rows 0-7 vs 8-15 in C tiles
    const int l15  = lane & 15;
    const int kSel = hi << 1;              // K offset 0/2 in 16x16x4 fragments
    const int rowBase = blockIdx.x * ROWS;
    const int colBase = wv * 128;
    const int i0 = (d - (T1c - 1)) > 0 ? (d - (T1c - 1)) : 0;

    // per-lane A-side geometry (row = rowBase + l15)
    const int rgA   = rowBase + l15;
    const int slotA = rgA >> 3;
    const int bbA   = rgA & 7;
    const int iiA   = i0 + slotA;
    const int jjA   = d - iiA;
    const int icA   = iclampi(iiA, 0, T0c - 1);
    const int jcA   = iclampi(jjA, 0, T1c - 1);
    const float* xBase   = x + (((size_t)bbA * T0c + icA) * T1c + jcA) * Ec;
    const float* sp0Base = S + (((size_t)(icA + 1) * (T1c + 2) + jcA) * Bc + bbA) * Hc;
    const float* sp1Base = S + (((size_t)icA * (T1c + 2) + (jcA + 1)) * Bc + bbA) * Hc;

    // per-lane B-side geometry (gate row = colBase + l15)
    const int nRow = colBase + l15;
    const float* wiRow = Wi + (size_t)nRow * Ec;        // Wi: (2048, 512)
    const float* wsRow = Ws + (size_t)nRow * (2 * Hc);  // Ws: (2048, 1024)

    v8f cg1[8], cg2[8];
#pragma unroll
    for (int t = 0; t < 8; ++t) { cg1[t] = (v8f)(0.0f); cg2[t] = (v8f)(0.0f); }

    // ---------- GEMM 1 (fused path only): igates_raw = xd . Wi^T, K=512 ----------
    if constexpr (!PRE) {
#pragma unroll 2
        for (int k = 0; k < Ec; k += 4) {
            __builtin_prefetch(xBase + k + 64, 0, 0);
            float2 a2 = *(const float2*)(xBase + k + kSel);
            v2f av; av.x = a2.x; av.y = a2.y;
#pragma unroll
            for (int t = 0; t < 8; ++t) {
                float2 b2 = *(const float2*)(wiRow + (size_t)t * 16 * Ec + k + kSel);
                v2f bv; bv.x = b2.x; bv.y = b2.y;
                cg1[t] = __builtin_amdgcn_wmma_f32_16x16x4_f32(
                    false, av, false, bv, (short)0, cg1[t], false, false);
            }
        }
    }

    // ---------- GEMM 2: sgates_raw = [sp0,sp1] . Ws^T, K=1024 ----------
#pragma unroll 2
    for (int k = 0; k < 2 * Hc; k += 4) {
        const float* aP = (k < Hc) ? (sp0Base + k) : (sp1Base + (k - Hc));
        float2 a2 = *(const float2*)(aP + kSel);
        v2f av; av.x = a2.x; av.y = a2.y;
#pragma unroll
        for (int t = 0; t < 8; ++t) {
            float2 b2 = *(const float2*)(wsRow + (size_t)t * 16 * (2 * Hc) + k + kSel);
            v2f bv; bv.x = b2.x; bv.y = b2.y;
            cg2[t] = __builtin_amdgcn_wmma_f32_16x16x4_f32(
                false, av, false, bv, (short)0, cg2[t], false, false);
        }
    }

    // ---------- row-wise LayerNorm stats over 2048 gates ----------
#pragma unroll
    for (int r = 0; r < 8; ++r) {
        float c2 = 0, e2 = 0;
#pragma unroll
        for (int t = 0; t < 8; ++t) { float v = cg2[t][r]; c2 += v; e2 += v * v; }
        c2 = halfReduce(c2); e2 = halfReduce(e2);
        if (l15 == 0) { int row = hi * 8 + r; sRed[wv][row][2] = c2; sRed[wv][row][3] = e2; }
        if constexpr (!PRE) {
            float a = 0, b = 0;
#pragma unroll
            for (int t = 0; t < 8; ++t) { float v = cg1[t][r]; a += v; b += v * v; }
            a = halfReduce(a); b = halfReduce(b);
            if (l15 == 0) { int row = hi * 8 + r; sRed[wv][row][0] = a; sRed[wv][row][1] = b; }
        }
    }
    __syncthreads();
    if (threadIdx.x < ROWS) {
        int row = threadIdx.x;
        float c2 = 0, e2 = 0;
        for (int w = 0; w < NWAVE; ++w) { c2 += sRed[w][row][2]; e2 += sRed[w][row][3]; }
        float m2 = c2 / (float)Gc;
        float v2 = e2 / (float)Gc - m2 * m2;
        sStats[row][2] = m2; sStats[row][3] = __frsqrt_rn(v2 + LN_EPS);
        if constexpr (!PRE) {
            float a = 0, b = 0;
            for (int w = 0; w < NWAVE; ++w) { a += sRed[w][row][0]; b += sRed[w][row][1]; }
            float m1 = a / (float)Gc;
            float v1 = b / (float)Gc - m1 * m1;
            sStats[row][0] = m1; sStats[row][1] = __frsqrt_rn(v1 + LN_EPS);
        }
    }
    __syncthreads();

    // per-row IG base offsets (PRE path)
    size_t igOff[8];
    if constexpr (PRE) {
#pragma unroll
        for (int r = 0; r < 8; ++r) {
            int row = hi * 8 + r;
            int rg = rowBase + row;
            int slot = rg >> 3, bb = rg & 7;
            int ii = i0 + slot, jj = d - ii;
            int ic = iclampi(ii, 0, T0c - 1), jc = iclampi(jj, 0, T1c - 1);
            igOff[r] = (((size_t)bb * T0c + ic) * T1c + jc) * (size_t)Gc;
        }
    }

    // ---------- affine LN + gate sum; stash slices i/n/l + sgates-n in LDS ----------
#pragma unroll
    for (int t = 0; t < 8; ++t) {
        int col = colBase + 16 * t + l15;
        float gs = g_s[col], bs = b_s[col];
        float gi = 0.0f, bi = 0.0f;
        if constexpr (!PRE) { gi = g_i[col]; bi = b_i[col]; }
#pragma unroll
        for (int r = 0; r < 8; ++r) {
            int row = hi * 8 + r;
            float g1v;
            if constexpr (PRE) g1v = IG[igOff[r] + col];
            else               g1v = (cg1[t][r] - sStats[row][0]) * sStats[row][1] * gi + bi;
            float g2v = (cg2[t][r] - sStats[row][2]) * sStats[row][3] * gs + bs;
            float gate = g1v + g2v;
            if (wv < 4) {
                cg1[t][r] = gate;                 // r-slice stays in registers
            } else {
                int slice = wv >> 2;              // 1=i_g, 2=n, 3=l
                int cIn = (wv & 3) * 128 + 16 * t + l15;
                sG[slice - 1][row][cIn] = gate;
                if (slice == 2) sSn[row][cIn] = g2v;
            }
        }
    }
    __syncthreads();

    // ---------- h computation (waves 0-3 own cols [0,512)) ----------
    float hs[8], hq[8];
#pragma unroll
    for (int r = 0; r < 8; ++r) { hs[r] = 0.0f; hq[r] = 0.0f; }
    if (wv < 4) {
#pragma unroll 1
        for (int r = 0; r < 8; ++r) {
            int row = hi * 8 + r;
            int rg = rowBase + row;
            int slot = rg >> 3, bb = rg & 7;
            int ii = i0 + slot, jj = d - ii;
            int ic = iclampi(ii, 0, T0c - 1), jc = iclampi(jj, 0, T1c - 1);
            const float* p0 = S + (((size_t)(ic + 1) * (T1c + 2) + jc) * Bc + bb) * Hc;
            const float* p1 = S + (((size_t)ic * (T1c + 2) + (jc + 1)) * Bc + bb) * Hc;
#pragma unroll
            for (int t = 0; t < 8; ++t) {
                int c = wv * 128 + 16 * t + l15;
                float rr = sigm(cg1[t][r]);
                float ig = sigm(sG[0][row][c]);
                float nr = sG[1][row][c];
                float ll = sigm(sG[2][row][c]);
                float sn = sSn[row][c];
                float sp0v = p0[c], sp1v = p1[c];
                float nn = tanhf(nr - rr * sn);
                float hh = nn + ig * (ll * sp0v + (1.0f - ll) * sp1v - nn);
                cg2[t][r] = hh;                   // reuse accum regs for h
                hs[r] += hh; hq[r] += hh * hh;
            }
        }
#pragma unroll
        for (int r = 0; r < 8; ++r) { hs[r] = halfReduce(hs[r]); hq[r] = halfReduce(hq[r]); }
        if (l15 == 0) {
#pragma unroll
            for (int r = 0; r < 8; ++r) {
                int row = hi * 8 + r;
                sHred[wv][row][0] = hs[r]; sHred[wv][row][1] = hq[r];
            }
        }
    }
    __syncthreads();
    if (threadIdx.x < ROWS) {
        int row = threadIdx.x;
        float a = 0, b = 0;
        for (int w = 0; w < 4; ++w) { a += sHred[w][row][0]; b += sHred[w][row][1]; }
        float m = a / (float)Hc;
        float v = b / (float)Hc - m * m;
        sHstats[row][0] = m; sHstats[row][1] = __frsqrt_rn(v + LN_EPS);
    }
    __syncthreads();

    // ---------- final LN(h) * mask, store to S / states_s / states_full ----------
    if (wv < 4) {
#pragma unroll 1
        for (int r = 0; r < 8; ++r) {
            int row = hi * 8 + r;
            int rg = rowBase + row;
            int slot = rg >> 3, bb = rg & 7;
            int ii = i0 + slot, jj = d - ii;
            bool valid = (ii < T0c) && (jj >= 0) && (jj < T1c);
            if (!valid) continue;
            float mval = masks[((size_t)bb * T0c + ii) * T1c + jj] ? 1.0f : 0.0f;
            float m = sHstats[row][0], rs = sHstats[row][1];
            float* sOut = S    + (((size_t)(ii + 1) * (T1c + 2) + (jj + 1)) * Bc + bb) * Hc;
            float* oS   = outS + (((size_t)bb * T0c + ii) * T1c + jj) * Hc;
            float* oF   = outF + (((size_t)(ii + 1) * (T1c + 1) + (T1c - 1 - jj)) * Bc + bb) * Hc;
#pragma unroll
            for (int t = 0; t < 8; ++t) {
                int c = wv * 128 + 16 * t + l15;
                float hn = (cg2[t][r] - m) * rs * g_h[c] + b_h[c];
                float o = hn * mval;
                sOut[c] = o; oS[c] = o; oF[c] = o;
            }
        }
    }
}

extern "C" void kernel_launch(void* const* d_in, const int* in_sizes, int n_in,
                              void* d_out, int out_size, void* d_ws, size_t ws_size,
                              hipStream_t stream) {
    const float* x  = (const float*)d_in[0];
    const unsigned char* masks = (const unsigned char*)d_in[1];
    const float* Wi = (const float*)d_in[2];
    const float* Ws = (const float*)d_in[3];
    const float* g_i = (const float*)d_in[4];
    const float* b_i = (const float*)d_in[5];
    const float* g_s = (const float*)d_in[6];
    const float* b_s = (const float*)d_in[7];
    const float* g_h = (const float*)d_in[8];
    const float* b_h = (const float*)d_in[9];

    float* S    = (float*)d_ws;                        // (T0+2, T1+2, B, H) fp32 state
    float* outS = (float*)d_out;                       // states_s  (B, T0, T1, H)
    float* outF = outS + (size_t)Bc * T0c * T1c * Hc;  // states_full (T0+1, T1+1, B, H)

    const size_t nS = (size_t)(T0c + 2) * (T1c + 2) * Bc * Hc;
    const size_t nRowsAll = (size_t)Bc * T0c * T1c;         // 32768
    const size_t nIG = nRowsAll * (size_t)Gc;               // 67M floats
    float* IG = S + nS;
    // Deterministic path choice: ws_size is fixed by the harness.
    const bool pre = ws_size >= (nS + nIG) * sizeof(float);

    gru2d_zero<<<2048, 256, 0, stream>>>(outS, (size_t)out_size, S, nS);

    dim3 blk(NWAVE * 32);                               // 512 threads = 16 wave32
    if (pre) {
        gru2d_igates<<<dim3((unsigned)(nRowsAll / ROWS)), blk, 0, stream>>>(x, Wi, g_i, b_i, IG);
    }

    dim3 grid(64 * Bc / ROWS);                          // 32 workgroups per diagonal
    for (int d = 0; d < T0c + T1c - 1; ++d) {
        if (pre) {
            gru2d_diag<true><<<grid, blk, 0, stream>>>(d, x, masks, Wi, Ws,
                                                       g_i, b_i, g_s, b_s, g_h, b_h,
                                                       IG, S, outS, outF);
        } else {
            gru2d_diag<false><<<grid, blk, 0, stream>>>(d, x, masks, Wi, Ws,
                                                        g_i, b_i, g_s, b_s, g_h, b_h,
                                                        IG, S, outS, outF);
        }
    }
}